// RoleBasedMask_42923903156437
// MI455X (gfx1250) — compile-verified
//
#include <hip/hip_runtime.h>
#include <hip/hip_bf16.h>
#include <stdint.h>

// Role-based attention mask:
//   out[b,0,i,j] = (allow[rq[b,i], rk[b,j]] > 0.5f) ? 0.0f : -1e9f
// B=4, Lq=Lk=4096, R=8.  Output 256MB f32 -> purely store-BW bound (~11.5us
// at 23.3 TB/s).  Strategy: async-stage roles_k row into LDS (gfx1250
// GLOBAL_LOAD_ASYNC_TO_LDS, ASYNCcnt), pack roles to bytes, build a per-rq
// 8-bit allow bitmap once per block, then stream float4 non-temporal stores.

typedef __attribute__((ext_vector_type(4))) float v4f;

#define LQ   4096
#define LK   4096
#define ROWS 16            // rows of the mask per block
#define TPB  256           // 8 waves (wave32)

__device__ __forceinline__ void async_load_b128_to_lds(unsigned lds_byte_off,
                                                       const void* gptr) {
    unsigned long long ga = (unsigned long long)(uintptr_t)gptr;
    // VDST = VGPR with LDS byte address, VADDR = 64-bit global address (GV mode)
    asm volatile("global_load_async_to_lds_b128 %0, %1, off"
                 :: "v"(lds_byte_off), "v"(ga) : "memory");
}

__device__ __forceinline__ void wait_asynccnt0() {
    asm volatile("s_wait_asynccnt 0x0" ::: "memory");
}

__global__ void __launch_bounds__(TPB)
role_mask_kernel(const int* __restrict__ roles_q,
                 const int* __restrict__ roles_k,
                 const float* __restrict__ allow,   // [8][8]
                 float* __restrict__ out)           // [B][1][LQ][LK]
{
    __shared__ unsigned lds_raw[LK];            // 16 KB staging: roles_k[b][*] (i32)
    __shared__ unsigned lds_pack[LK / 4];       // 4 KB: 4 roles (bytes) per word
    __shared__ unsigned lds_bm[2];              // 64-bit allow bitmap
    __shared__ int      lds_rq[ROWS];           // this block's query roles

    const int tid = threadIdx.x;
    const int b  = blockIdx.x >> 8;             // blockIdx.x / (LQ/ROWS)
    const int rb = blockIdx.x & 255;            // row-block within batch

    // ---- Phase 0: async copy roles_k[b][0..4095] (16 KB) into LDS ----------
    // 4 issues x 256 threads x 16B = 16 KB, ASYNCcnt-tracked.
    const char* krow = (const char*)(roles_k + (size_t)b * LK);
    #pragma unroll
    for (int it = 0; it < 4; ++it) {
        unsigned byte_off = (unsigned)(it * 4096 + tid * 16);
        unsigned lds_off  = (unsigned)(uintptr_t)lds_raw + byte_off;
        async_load_b128_to_lds(lds_off, krow + byte_off);
    }

    if (tid < 2) lds_bm[tid] = 0u;

    wait_asynccnt0();          // my wave's async loads complete
    __syncthreads();           // all waves' loads + bitmap init visible

    // ---- Phase 1: allow bitmap, rq preload, pack roles_k to bytes ----------
    if (tid < 64) {
        unsigned bit = (allow[tid] > 0.5f) ? 1u : 0u;
        atomicOr(&lds_bm[tid >> 5], bit << (tid & 31));
    }
    if (tid < ROWS) {
        lds_rq[tid] = roles_q[(size_t)b * LQ + rb * ROWS + tid];
    }
    const uint4* raw4 = (const uint4*)lds_raw;
    #pragma unroll
    for (int p = 0; p < 4; ++p) {
        int idx = p * TPB + tid;                // packed word 0..1023
        uint4 r = raw4[idx];                    // ds_load_b128, conflict-free
        lds_pack[idx] = (r.x & 7u) | ((r.y & 7u) << 8) |
                        ((r.z & 7u) << 16) | ((r.w & 7u) << 24);
    }
    __syncthreads();

    const unsigned bmlo = lds_bm[0];
    const unsigned bmhi = lds_bm[1];
    const float NEGV = -1000000000.0f;

    // ---- Phase 2: stream 16 rows x 4096 cols of float4 NT stores -----------
    for (int r = 0; r < ROWS; ++r) {
        const int rq = lds_rq[r] & 7;
        // 8-bit slice of the bitmap for this query role
        const unsigned bits8 = (rq < 4) ? (bmlo >> (rq * 8))
                                        : (bmhi >> ((rq - 4) * 8));
        float* orow = out + ((size_t)(b * LQ + rb * ROWS + r)) * (size_t)LK;
        #pragma unroll
        for (int j4 = tid; j4 < LK / 4; j4 += TPB) {
            const unsigned pk = lds_pack[j4];   // 4 key roles, one byte each
            v4f v;
            v.x = ((bits8 >> ( pk        & 7u)) & 1u) ? 0.0f : NEGV;
            v.y = ((bits8 >> ((pk >>  8) & 7u)) & 1u) ? 0.0f : NEGV;
            v.z = ((bits8 >> ((pk >> 16) & 7u)) & 1u) ? 0.0f : NEGV;
            v.w = ((bits8 >> ((pk >> 24) & 7u)) & 1u) ? 0.0f : NEGV;
            // 256MB streamed once: non-temporal (global_store_b128 th:NT)
            __builtin_nontemporal_store(v, (v4f*)(orow + (size_t)j4 * 4));
        }
    }
}

extern "C" void kernel_launch(void* const* d_in, const int* in_sizes, int n_in,
                              void* d_out, int out_size, void* d_ws, size_t ws_size,
                              hipStream_t stream) {
    const int*   roles_q = (const int*)d_in[0];
    const int*   roles_k = (const int*)d_in[1];
    const float* allow   = (const float*)d_in[2];
    float*       out     = (float*)d_out;

    const int B = in_sizes[0] / LQ;             // = 4
    dim3 grid((unsigned)(B * (LQ / ROWS)));     // 1024 blocks
    dim3 block(TPB);
    role_mask_kernel<<<grid, block, 0, stream>>>(roles_q, roles_k, allow, out);
}